// ICR_83923660963974
// MI455X (gfx1250) — compile-verified
//
#include <hip/hip_runtime.h>
#include <math.h>

#define BS   8
#define P    2048
#define CIN  512
#define NCLS 3
#define EPS_ 1e-7f
#define I_T  0.5f
#define NEG_ -1000000000.0f

typedef __attribute__((ext_vector_type(2))) float v2f;
typedef __attribute__((ext_vector_type(8))) float v8f;

// ---------------------------------------------------------------------------
// Kernel A: xr = inputs @ fc_w^T + fc_b ; logit = softmax(xr, axis=-1)
// V_WMMA_F32_16X16X4_F32, fp32 exact path. One wave = one 16-row tile,
// 8 waves/block. B panel is PADDED to 16 rows in LDS (rows 4..15 = 0) so the
// inner-loop ds_load is unconditional: no EXEC save/restore per iteration.
// ---------------------------------------------------------------------------
__global__ __launch_bounds__(256) void icr_gemm_softmax(
    const float* __restrict__ inputs,   // (BS*P, CIN)
    const float* __restrict__ fc_w,     // (4, CIN)
    const float* __restrict__ fc_b,     // (4,)
    float* __restrict__ logit)          // (BS*P, 4)
{
  __shared__ float s_w[16 * CIN];       // 32 KB padded B panel (rows 4..15 = 0)
  __shared__ float s_xr[8][16][4];      // 2 KB per-wave tile results

  // cooperative fill: rows 0..3 <- fc_w, rows 4..15 <- 0   (2048 float4)
  {
    const float4* wsrc = reinterpret_cast<const float4*>(fc_w);
    float4* wdst = reinterpret_cast<float4*>(s_w);
    const float4 z4 = make_float4(0.f, 0.f, 0.f, 0.f);
    #pragma unroll
    for (int t = (int)threadIdx.x; t < (16 * CIN) / 4; t += 256)
      wdst[t] = (t < (4 * CIN) / 4) ? wsrc[t] : z4;
  }
  __syncthreads();

  const int wave = threadIdx.x >> 5;
  const int lane = threadIdx.x & 31;
  const int half = lane >> 4;           // half-wave selects K pair (0,1) vs (2,3)
  const int n    = lane & 15;           // M index for A, N index for B/C/D

  const int tile = blockIdx.x * 8 + wave;      // 1024 tiles of 16 rows
  const int row0 = tile * 16;
  const float* arow = inputs + (size_t)(row0 + n) * CIN;
  const float* brow = s_w + n * CIN;    // padded: valid for all 16 lanes

  v8f acc = {};
  #pragma unroll 8
  for (int k = 0; k < CIN; k += 4) {
    const int ka = k + 2 * half;
    // A: 16x4 fp32 — lanes 0-15 hold K=k,k+1 of row M=lane&15; lanes 16-31 K=k+2,k+3
    const float2 av = *reinterpret_cast<const float2*>(arow + ka);
    // B: 4x16 fp32 (padded to 16x16) — row K striped across lanes (N)
    const float2 bv = *reinterpret_cast<const float2*>(brow + ka);
    v2f a; a.x = av.x; a.y = av.y;
    v2f b; b.x = bv.x; b.y = bv.y;
    acc = __builtin_amdgcn_wmma_f32_16x16x4_f32(
        /*neg_a=*/false, a, /*neg_b=*/false, b,
        /*c_mod=*/(short)0, acc, /*reuse_a=*/false, /*reuse_b=*/false);
  }

  // D layout: VGPR v holds row M = v + 8*half, column N = lane&15
  if (n < 4) {
    const float bias = fc_b[n];
    #pragma unroll
    for (int v = 0; v < 8; ++v) s_xr[wave][v + 8 * half][n] = acc[v] + bias;
  }
  __syncthreads();

  // softmax: one thread per row (128 rows/block)
  if (threadIdx.x < 128) {
    const int w8 = threadIdx.x >> 4;
    const int r  = threadIdx.x & 15;
    const float x0 = s_xr[w8][r][0], x1 = s_xr[w8][r][1];
    const float x2 = s_xr[w8][r][2], x3 = s_xr[w8][r][3];
    const float mx = fmaxf(fmaxf(x0, x1), fmaxf(x2, x3));
    const float e0 = __expf(x0 - mx), e1 = __expf(x1 - mx);
    const float e2 = __expf(x2 - mx), e3 = __expf(x3 - mx);
    const float inv = 1.0f / (e0 + e1 + e2 + e3);
    const int grow = (blockIdx.x * 8 + w8) * 16 + r;
    reinterpret_cast<float4*>(logit)[grow] =
        make_float4(e0 * inv, e1 * inv, e2 * inv, e3 * inv);
  }
}

// ---------------------------------------------------------------------------
// Kernel B0: per (batch,class) selection mask.
//   count = #(score>0.5); argmax with first-occurrence tiebreak;
//   sel[b][c][j] = valid && (count>=2 ? score>0.5 : j==argmax)
// Block 0 also zeroes imbalance[4] and the loss scalar.
// ---------------------------------------------------------------------------
__global__ __launch_bounds__(256) void icr_select(
    const float* __restrict__ pre_score,   // (BS,P,4)
    const float* __restrict__ labels,      // (BS,4)
    unsigned char* __restrict__ sel,       // (BS,NCLS,P)
    float* __restrict__ imb,               // (4,)
    float* __restrict__ loss)              // scalar
{
  if (blockIdx.x == 0 && threadIdx.x < 5) {
    if (threadIdx.x < 4) imb[threadIdx.x] = 0.0f;
    else                 *loss = 0.0f;
  }

  const int b = blockIdx.x / NCLS;
  const int c = blockIdx.x % NCLS;

  __shared__ int   s_cnt[256];
  __shared__ float s_max[256];
  __shared__ int   s_idx[256];

  int cnt = 0; float mv = -INFINITY; int mi = 0x7fffffff;
  float vals[P / 256];
  for (int t = 0; t < P / 256; ++t) {
    const int j = t * 256 + threadIdx.x;
    const float v = pre_score[((size_t)b * P + j) * 4 + c];
    vals[t] = v;
    if (v > 0.5f) ++cnt;
    if (v > mv) { mv = v; mi = j; }      // strict > keeps first occurrence
  }
  s_cnt[threadIdx.x] = cnt; s_max[threadIdx.x] = mv; s_idx[threadIdx.x] = mi;
  __syncthreads();
  for (int s = 128; s > 0; s >>= 1) {
    if ((int)threadIdx.x < s) {
      s_cnt[threadIdx.x] += s_cnt[threadIdx.x + s];
      const float ov = s_max[threadIdx.x + s];
      const int   oi = s_idx[threadIdx.x + s];
      if (ov > s_max[threadIdx.x] ||
          (ov == s_max[threadIdx.x] && oi < s_idx[threadIdx.x])) {
        s_max[threadIdx.x] = ov; s_idx[threadIdx.x] = oi;
      }
    }
    __syncthreads();
  }
  const int  count = s_cnt[0];
  const int  argj  = s_idx[0];
  const bool valid = labels[b * 4 + c] > 0.0f;

  for (int t = 0; t < P / 256; ++t) {
    const int j = t * 256 + threadIdx.x;
    const bool s = valid && ((count >= 2) ? (vals[t] > 0.5f) : (j == argj));
    sel[((size_t)b * NCLS + c) * P + j] = s ? 1 : 0;
  }
}

// ---------------------------------------------------------------------------
// Kernel B1: IoU assignment. One block = 256 proposals of one batch.
// ROIs (32KB) + sel masks (6KB) staged in LDS; each thread scans all j.
// ---------------------------------------------------------------------------
__global__ __launch_bounds__(256) void icr_assign(
    const float* __restrict__ rois,          // (BS,P,4)
    const float* __restrict__ pre_score,     // (BS,P,4)
    const unsigned char* __restrict__ sel,   // (BS,NCLS,P)
    float* __restrict__ yk,                  // (BS,P,4)
    float* __restrict__ wv,                  // (BS,P)
    float* __restrict__ imb)                 // (4,)
{
  __shared__ float4 s_roi[P];                               // 32 KB
  __shared__ __align__(16) unsigned char s_sel[NCLS][P];    // 6 KB
  __shared__ float s_red[256];

  const int b = blockIdx.x >> 3;
  const int i = ((blockIdx.x & 7) << 8) + threadIdx.x;

  const float4* rsrc = reinterpret_cast<const float4*>(rois + (size_t)b * P * 4);
  for (int t = (int)threadIdx.x; t < P; t += 256) s_roi[t] = rsrc[t];
  const int4* ssrc = reinterpret_cast<const int4*>(sel + (size_t)b * NCLS * P);
  int4* sdst = reinterpret_cast<int4*>(&s_sel[0][0]);
  for (int t = (int)threadIdx.x; t < (NCLS * P) / 16; t += 256) sdst[t] = ssrc[t];
  __syncthreads();

  const float4 bi = s_roi[i];
  const float areai = (bi.z - bi.x) * (bi.w - bi.y);

  float pcmax[NCLS] = {NEG_, NEG_, NEG_};
  int   pcidx[NCLS] = {0, 0, 0};

  for (int j = 0; j < P; ++j) {
    const float4 bj = s_roi[j];                   // broadcast read
    const float areaj = (bj.z - bj.x) * (bj.w - bj.y);
    const float ltx = fmaxf(bi.x, bj.x);
    const float lty = fmaxf(bi.y, bj.y);
    const float rbx = fminf(bi.z, bj.z);
    const float rby = fminf(bi.w, bj.w);
    const float iw = fmaxf(rbx - ltx, 0.0f);
    const float ih = fmaxf(rby - lty, 0.0f);
    const float inter = iw * ih;
    const float iou = inter / (areai + areaj - inter);
    #pragma unroll
    for (int c = 0; c < NCLS; ++c) {
      if (s_sel[c][j] && iou > pcmax[c]) { pcmax[c] = iou; pcidx[c] = j; }
    }
  }

  float y[4];
  bool any = false;
  #pragma unroll
  for (int c = 0; c < NCLS; ++c) {
    const bool h = pcmax[c] > I_T;
    y[c] = h ? 1.0f : 0.0f;
    any = any || h;
  }
  y[3] = any ? 0.0f : 1.0f;

  // flat argmax over (c,j): lowest c, then lowest j on ties
  float bestv = -INFINITY; int bestc = 0, bestj = 0;
  #pragma unroll
  for (int c = 0; c < NCLS; ++c)
    if (pcmax[c] > bestv) { bestv = pcmax[c]; bestc = c; bestj = pcidx[c]; }

  float wi = 1.0f;
  if (bestv > -1.0f)
    wi = pre_score[((size_t)b * P + bestj) * 4 + bestc];

  const size_t gi = (size_t)b * P + i;
  reinterpret_cast<float4*>(yk)[gi] = make_float4(y[0], y[1], y[2], y[3]);
  wv[gi] = wi;

  // block-reduce y into imbalance
  #pragma unroll
  for (int k = 0; k < 4; ++k) {
    s_red[threadIdx.x] = y[k];
    __syncthreads();
    for (int s = 128; s > 0; s >>= 1) {
      if ((int)threadIdx.x < s) s_red[threadIdx.x] += s_red[threadIdx.x + s];
      __syncthreads();
    }
    if (threadIdx.x == 0) atomicAdd(&imb[k], s_red[0]);
    __syncthreads();
  }
}

// ---------------------------------------------------------------------------
// Kernel C: focal loss + weighting + global scalar reduction.
// ---------------------------------------------------------------------------
__global__ __launch_bounds__(256) void icr_loss(
    const float* __restrict__ logit,   // (BS,P,4)  == d_out
    const float* __restrict__ yk,      // (BS,P,4)
    const float* __restrict__ wv,      // (BS,P)
    const float* __restrict__ labels,  // (BS,4)
    const float* __restrict__ imb,     // (4,)
    float* __restrict__ loss)          // scalar -> d_out[BS*P*4]
{
  __shared__ float s_red[256];
  const int b = blockIdx.x >> 3;
  const int i = ((blockIdx.x & 7) << 8) + threadIdx.x;
  const size_t gi = (size_t)b * P + i;

  const float4 lg = reinterpret_cast<const float4*>(logit)[gi];
  const float4 y4 = reinterpret_cast<const float4*>(yk)[gi];
  const float  w  = wv[gi];
  const float l[4] = {lg.x, lg.y, lg.z, lg.w};
  const float yy[4] = {y4.x, y4.y, y4.z, y4.w};

  float acc = 0.0f;
  #pragma unroll
  for (int k = 0; k < 4; ++k) {
    const float pk  = fminf(fmaxf(l[k], EPS_), 1.0f - EPS_);
    const float omp = 1.0f - pk;
    const float fl  = -yy[k] * __logf(pk) * omp * omp;   // gamma = 2
    const float lab = (k == 3) ? 1.0f : labels[b * 4 + k];
    const float w_  = 10.0f * __expf(l[k]) * (1.0f - lab) + lab;
    acc += w * fl / (imb[k] + 1e-7f) * w_;
  }

  s_red[threadIdx.x] = acc;
  __syncthreads();
  for (int s = 128; s > 0; s >>= 1) {
    if ((int)threadIdx.x < s) s_red[threadIdx.x] += s_red[threadIdx.x + s];
    __syncthreads();
  }
  if (threadIdx.x == 0) atomicAdd(loss, s_red[0] / (float)BS);
}

// ---------------------------------------------------------------------------
extern "C" void kernel_launch(void* const* d_in, const int* in_sizes, int n_in,
                              void* d_out, int out_size, void* d_ws, size_t ws_size,
                              hipStream_t stream) {
  const float* inputs    = (const float*)d_in[0];   // (BS*P, CIN)
  const float* pre_score = (const float*)d_in[1];   // (BS,P,4)
  const float* labels    = (const float*)d_in[2];   // (BS,4)
  const float* rois      = (const float*)d_in[3];   // (BS,P,4)
  const float* fc_w      = (const float*)d_in[4];   // (4,CIN)
  const float* fc_b      = (const float*)d_in[5];   // (4,)

  float* out   = (float*)d_out;
  float* logit = out;                 // first BS*P*4 floats
  float* loss  = out + (size_t)BS * P * 4;

  // workspace layout
  char* ws = (char*)d_ws;
  unsigned char* sel = (unsigned char*)ws;                 // BS*NCLS*P bytes
  size_t off = (size_t)BS * NCLS * P;                      // 49152
  off = (off + 255) & ~(size_t)255;
  float* yk  = (float*)(ws + off);  off += (size_t)BS * P * 4 * sizeof(float);
  float* wv  = (float*)(ws + off);  off += (size_t)BS * P * sizeof(float);
  float* imb = (float*)(ws + off);

  // A: WMMA GEMM + softmax -> logit
  icr_gemm_softmax<<<(BS * P) / 128, 256, 0, stream>>>(inputs, fc_w, fc_b, logit);
  // B0: selection masks (+ zero imbalance & loss)
  icr_select<<<BS * NCLS, 256, 0, stream>>>(pre_score, labels, sel, imb, loss);
  // B1: IoU assignment -> y_k, w, imbalance
  icr_assign<<<BS * (P / 256), 256, 0, stream>>>(rois, pre_score, sel, yk, wv, imb);
  // C: loss reduction -> d_out[BS*P*4]
  icr_loss<<<BS * (P / 256), 256, 0, stream>>>(logit, yk, wv, labels, imb, loss);
}